// SAM_64183991271998
// MI455X (gfx1250) — compile-verified
//
#include <hip/hip_runtime.h>

// ---- problem constants (from reference) ----
#define BATCH 4
#define CIN   256
#define H     56
#define W     56
#define HW    (H*W)          // 3136
#define CR    32
#define COUT  256
#define SP    8
#define HN    32             // COUT/SP
#define KS    7
#define PADR  3
#define NK    49
#define KCHUNK 7             // offsets per wave in logits kernel (49 = 7*7)
#define EPSV  1e-5f

typedef float v2f __attribute__((ext_vector_type(2)));
typedef float v8f __attribute__((ext_vector_type(8)));

static __device__ __forceinline__ v8f wmma_f32(v2f a, v2f b, v8f c) {
  // D(16x16) = A(16x4) x B(4x16) + C  — full fp32 matrix op on CDNA5
  return __builtin_amdgcn_wmma_f32_16x16x4_f32(false, a, false, b, (short)0, c, false, false);
}

static __device__ __forceinline__ int reflect_idx(int p, int n) {
  // jnp.pad mode='reflect' (no edge repeat)
  if (p < 0) p = -p;
  else if (p >= n) p = 2*n - 2 - p;
  return p;
}

// ---------------------------------------------------------------------------
// Kernel 0: fold BN constants + identity epilogue arrays.
//   inv1 = g1/sqrt(v1+eps); beta1 = b1 - m1*inv1
//   cw1s[o][k] = cw1[o][k] * inv2[o]; beta2 = b2 - m2*inv2
//   sc3[o]=1, sh3[o]=0  (branch-free conv epilogue for x2/x3 passes)
// ---------------------------------------------------------------------------
__global__ __launch_bounds__(1024) void prep_kernel(
    const float* __restrict__ bn1_g, const float* __restrict__ bn1_b,
    const float* __restrict__ bn1_m, const float* __restrict__ bn1_v,
    const float* __restrict__ bn2_g, const float* __restrict__ bn2_b,
    const float* __restrict__ bn2_m, const float* __restrict__ bn2_v,
    const float* __restrict__ cw1,
    float* __restrict__ inv1, float* __restrict__ beta1,
    float* __restrict__ beta2, float* __restrict__ cw1s,
    float* __restrict__ sc3, float* __restrict__ sh3)
{
  int t = threadIdx.x;
  if (t < CR) {
    float i1 = bn1_g[t] * rsqrtf(bn1_v[t] + EPSV);
    inv1[t]  = i1;
    beta1[t] = bn1_b[t] - bn1_m[t] * i1;
    float i2 = bn2_g[t] * rsqrtf(bn2_v[t] + EPSV);
    beta2[t] = bn2_b[t] - bn2_m[t] * i2;
  }
  if (t < CR*CR) {
    int o = t / CR;
    float i2 = bn2_g[o] * rsqrtf(bn2_v[o] + EPSV);
    cw1s[t] = cw1[t] * i2;
  }
  if (t < COUT) { sc3[t] = 1.f; sh3[t] = 0.f; }
}

// ---------------------------------------------------------------------------
// Kernel 1: 1x1 conv as GEMM  out[b,o,i] = ((sum_c w[o,c]*x[b,c,i]) + bias[o])
//                                          * scale[o] + shift[o]
// One wave computes a 16(out)x16(pixel) tile via f32 WMMA, K in steps of 4.
// scale/shift are always valid arrays (identity for the plain pass).
// ---------------------------------------------------------------------------
__global__ __launch_bounds__(256) void conv1x1_wmma(
    const float* __restrict__ x, const float* __restrict__ w,
    const float* __restrict__ bias,
    const float* __restrict__ scale, const float* __restrict__ shift,
    float* __restrict__ out, int O)
{
  const int wave = threadIdx.x >> 5;
  const int lane = threadIdx.x & 31;
  const int gw   = blockIdx.x * 8 + wave;
  const int nOT  = O >> 4;
  const int nTiles = nOT * ((BATCH*HW) >> 4);
  if (gw >= nTiles) return;                  // wave-uniform
  const int oT = gw % nOT, pT = gw / nOT;
  const int ln = lane & 15, half = lane >> 4;
  const int o0 = oT << 4;
  const int p  = (pT << 4) + ln;             // HW%16==0 -> tile stays in batch
  const int bb = p / HW, ii = p - bb * HW;

  const float* xb = x + (size_t)bb * CIN * HW + ii;       // column pixel base
  const float* wr = w + (size_t)(o0 + ln) * CIN;          // A row base

  v8f acc;
  #pragma unroll
  for (int r = 0; r < 8; ++r) acc[r] = 0.f;

  #pragma unroll 8
  for (int kk = 0; kk < CIN; kk += 4) {
    const int ka = kk + 2*half;              // A: VGPR0=K(+2*half), VGPR1=+1
    v2f a; a.x = wr[ka];                 a.y = wr[ka + 1];
    v2f b; b.x = xb[(size_t)ka * HW];    b.y = xb[(size_t)(ka + 1) * HW];
    acc = wmma_f32(a, b, acc);
  }

  #pragma unroll
  for (int r = 0; r < 8; ++r) {
    const int m = 8*half + r;                // C/D: row = r + 8*half
    const int o = o0 + m;
    const float v = (acc[r] + bias[o]) * scale[o] + shift[o];
    out[((size_t)bb * O + o) * HW + ii] = v;
  }
}

// ---------------------------------------------------------------------------
// Kernel 2: logits[b,hn,k,i].
// One wave per (16-pixel tile, 7-offset chunk) -> 5488 waves.
//   Y  = relu(x1' - x2'shift)        built directly in B-operand layout
//   Z  = cw1s @ Y + beta2  (WMMA)    -> relu -> LDS relayout
//   L  = cw2  @ Z + cw2_b  (WMMA)    -> store
// ---------------------------------------------------------------------------
__global__ __launch_bounds__(256) void logits_kernel(
    const float* __restrict__ x1, const float* __restrict__ x2,
    const float* __restrict__ cw1s, const float* __restrict__ beta2,
    const float* __restrict__ cw2, const float* __restrict__ cw2b,
    float* __restrict__ logits)
{
  __shared__ float zbuf[8][CR][16];

  const int wave = threadIdx.x >> 5;
  const int lane = threadIdx.x & 31;
  const int gw   = blockIdx.x * 8 + wave;    // grid sized exactly: no idle waves
  const int ln = lane & 15, half = lane >> 4;

  const int pt    = gw / KCHUNK;             // pixel tile
  const int k0    = (gw - pt * KCHUNK) * KCHUNK;  // first offset of this chunk
  const int kEnd  = k0 + KCHUNK;

  const int p  = (pt << 4) + ln;
  const int bb = p / HW, ii = p - bb * HW;
  const int yy = ii / W, xx = ii - yy * W;

  // Preload weight A-operands for both M-tiles and all 8 K-steps.
  v2f cw1A[2][8], cw2A[2][8];
  #pragma unroll
  for (int t = 0; t < 2; ++t) {
    #pragma unroll
    for (int j = 0; j < 8; ++j) {
      const int c0  = 4*j + 2*half;
      const int row = 16*t + ln;
      cw1A[t][j].x = cw1s[row*CR + c0]; cw1A[t][j].y = cw1s[row*CR + c0 + 1];
      cw2A[t][j].x = cw2 [row*CR + c0]; cw2A[t][j].y = cw2 [row*CR + c0 + 1];
    }
  }

  // Preload x1' (BN1 already folded) for this lane's pixel / channel slots.
  float y1v[16];
  #pragma unroll
  for (int j = 0; j < 8; ++j) {
    const int c0 = 4*j + 2*half;
    y1v[2*j]   = x1[((size_t)bb*CR + c0    ) * HW + ii];
    y1v[2*j+1] = x1[((size_t)bb*CR + c0 + 1) * HW + ii];
  }

  // Per-lane C-init constants in D layout (channel = r + 8*half + 16*t).
  float bz[16], lb[16];
  #pragma unroll
  for (int t = 0; t < 2; ++t) {
    #pragma unroll
    for (int r = 0; r < 8; ++r) {
      const int m = 16*t + 8*half + r;
      bz[8*t + r] = beta2[m];
      lb[8*t + r] = cw2b[m];
    }
  }

  const float* x2b = x2 + (size_t)bb * CR * HW;
  float* lgb = logits + (size_t)(bb * HN) * NK * HW + ii;

  for (int k = k0; k < kEnd; ++k) {
    const int dy = k / KS - PADR, dx = k % KS - PADR;
    const int isrc = reflect_idx(yy + dy, H) * W + reflect_idx(xx + dx, W);

    // prefetch next offset's x2 line for this lane's first channel
    if (k + 1 < kEnd) {
      const int dy2 = (k + 1) / KS - PADR, dx2 = (k + 1) % KS - PADR;
      const int is2 = reflect_idx(yy + dy2, H) * W + reflect_idx(xx + dx2, W);
      __builtin_prefetch(&x2b[(size_t)(2*half) * HW + is2], 0, 3);
    }

    // ---- GEMM1: Z = cw1s @ relu(x1' - x2'shift) + beta2 ----
    v8f z0, z1;
    #pragma unroll
    for (int r = 0; r < 8; ++r) { z0[r] = bz[r]; z1[r] = bz[8 + r]; }

    #pragma unroll
    for (int j = 0; j < 8; ++j) {
      const int c0 = 4*j + 2*half;
      v2f yv;
      yv.x = fmaxf(y1v[2*j]   - x2b[(size_t)(c0    ) * HW + isrc], 0.f);
      yv.y = fmaxf(y1v[2*j+1] - x2b[(size_t)(c0 + 1) * HW + isrc], 0.f);
      z0 = wmma_f32(cw1A[0][j], yv, z0);
      z1 = wmma_f32(cw1A[1][j], yv, z1);
    }

    // ---- relayout relu(Z) through LDS: D layout -> row-major [ch][pix] ----
    __syncthreads();   // WAR vs previous iteration's reads (uniform trip count)
    #pragma unroll
    for (int r = 0; r < 8; ++r) {
      zbuf[wave][ 8*half + r     ][ln] = fmaxf(z0[r], 0.f);
      zbuf[wave][16 + 8*half + r ][ln] = fmaxf(z1[r], 0.f);
    }
    __syncthreads();

    // ---- GEMM2: L = cw2 @ relu(Z) + cw2_b ----
    v8f l0, l1;
    #pragma unroll
    for (int r = 0; r < 8; ++r) { l0[r] = lb[r]; l1[r] = lb[8 + r]; }

    #pragma unroll
    for (int j = 0; j < 8; ++j) {
      const int c0 = 4*j + 2*half;
      v2f zv; zv.x = zbuf[wave][c0][ln]; zv.y = zbuf[wave][c0 + 1][ln];
      l0 = wmma_f32(cw2A[0][j], zv, l0);
      l1 = wmma_f32(cw2A[1][j], zv, l1);
    }

    // ---- store logits[b, m, k, i] ----
    float* lgp = lgb + (size_t)k * HW;
    #pragma unroll
    for (int r = 0; r < 8; ++r) {
      const int m0 = 8*half + r;
      lgp[(size_t)(m0     ) * NK * HW] = l0[r];
      lgp[(size_t)(m0 + 16) * NK * HW] = l1[r];
    }
  }
}

// ---------------------------------------------------------------------------
// Kernel 3: softmax over the 49 offsets + weighted aggregation of x3.
// One thread per (b, head, pixel). Logit loads are coalesced across lanes.
// ---------------------------------------------------------------------------
__global__ __launch_bounds__(256) void softmax_agg_kernel(
    const float* __restrict__ logits, const float* __restrict__ x3,
    float* __restrict__ out)
{
  const int t = blockIdx.x * 256 + threadIdx.x;   // grid exact: B*HN*HW threads
  const int b  = t / (HN*HW);
  const int r0 = t - b * HN * HW;
  const int hn = r0 / HW;
  const int ii = r0 - hn * HW;
  const int yy = ii / W, xx = ii - yy * W;

  const float* lp = logits + (size_t)(b * HN + hn) * NK * HW + ii;
  float lg[NK];
  float mx = -1e30f;
  #pragma unroll
  for (int k = 0; k < NK; ++k) { lg[k] = lp[(size_t)k * HW]; mx = fmaxf(mx, lg[k]); }
  float s = 0.f;
  #pragma unroll
  for (int k = 0; k < NK; ++k) { lg[k] = __expf(lg[k] - mx); s += lg[k]; }
  const float inv = 1.f / s;

  float acc[SP];
  #pragma unroll
  for (int sp = 0; sp < SP; ++sp) acc[sp] = 0.f;

  const float* x3p = x3 + ((size_t)b * COUT + hn * SP) * HW;
  #pragma unroll
  for (int k = 0; k < NK; ++k) {
    const int dy = k / KS - PADR, dx = k % KS - PADR;
    const int isrc = reflect_idx(yy + dy, H) * W + reflect_idx(xx + dx, W);
    const float wk = lg[k] * inv;
    #pragma unroll
    for (int sp = 0; sp < SP; ++sp)
      acc[sp] += x3p[(size_t)sp * HW + isrc] * wk;
  }

  float* op = out + ((size_t)b * COUT + hn * SP) * HW + ii;
  #pragma unroll
  for (int sp = 0; sp < SP; ++sp) op[(size_t)sp * HW] = acc[sp];
}

// ---------------------------------------------------------------------------
// launch
// ---------------------------------------------------------------------------
extern "C" void kernel_launch(void* const* d_in, const int* in_sizes, int n_in,
                              void* d_out, int out_size, void* d_ws, size_t ws_size,
                              hipStream_t stream) {
  const float* x     = (const float*)d_in[0];
  const float* w1c   = (const float*)d_in[1];
  const float* b1c   = (const float*)d_in[2];
  const float* w2c   = (const float*)d_in[3];
  const float* b2c   = (const float*)d_in[4];
  const float* w3c   = (const float*)d_in[5];
  const float* b3c   = (const float*)d_in[6];
  const float* bn1_g = (const float*)d_in[7];
  const float* bn1_b = (const float*)d_in[8];
  const float* bn1_m = (const float*)d_in[9];
  const float* bn1_v = (const float*)d_in[10];
  const float* cw1   = (const float*)d_in[11];
  const float* bn2_g = (const float*)d_in[12];
  const float* bn2_b = (const float*)d_in[13];
  const float* bn2_m = (const float*)d_in[14];
  const float* bn2_v = (const float*)d_in[15];
  const float* cw2   = (const float*)d_in[16];
  const float* cw2b  = (const float*)d_in[17];
  float* out = (float*)d_out;

  // workspace layout (floats)
  float* ws = (float*)d_ws;
  const size_t n_x12 = (size_t)BATCH * CR * HW;      // 401408
  const size_t n_x3  = (size_t)BATCH * COUT * HW;    // 3211264
  const size_t n_lg  = (size_t)BATCH * HN * NK * HW; // 19668992
  float* x1w   = ws;
  float* x2w   = x1w + n_x12;
  float* x3w   = x2w + n_x12;
  float* lgw   = x3w + n_x3;
  float* inv1  = lgw + n_lg;
  float* beta1 = inv1 + CR;
  float* beta2 = beta1 + CR;
  float* cw1s  = beta2 + CR;
  float* sc3   = cw1s + CR*CR;
  float* sh3   = sc3 + COUT;

  // 0) fold BN constants + identity epilogue arrays
  prep_kernel<<<1, 1024, 0, stream>>>(bn1_g, bn1_b, bn1_m, bn1_v,
                                      bn2_g, bn2_b, bn2_m, bn2_v,
                                      cw1, inv1, beta1, beta2, cw1s, sc3, sh3);

  // 1) the three 1x1 convs (BN1 folded into x1/x2; branch-free epilogue)
  const int pixTiles = (BATCH * HW) / 16;            // 784
  {
    int nT = (CR / 16) * pixTiles;                   // 1568
    conv1x1_wmma<<<nT / 8, 256, 0, stream>>>(x, w1c, b1c, inv1, beta1, x1w, CR);
    conv1x1_wmma<<<nT / 8, 256, 0, stream>>>(x, w2c, b2c, inv1, sh3,   x2w, CR);
  }
  {
    int nT = (COUT / 16) * pixTiles;                 // 12544
    conv1x1_wmma<<<nT / 8, 256, 0, stream>>>(x, w3c, b3c, sc3, sh3, x3w, COUT);
  }

  // 2) attention logits: (pixel tile, offset chunk) -> 5488 waves
  {
    int nWaves = pixTiles * KCHUNK;                  // 5488
    logits_kernel<<<nWaves / 8, 256, 0, stream>>>(x1w, x2w, cw1s, beta2, cw2, cw2b, lgw);
  }

  // 3) softmax over offsets + aggregation of x3 neighborhoods
  {
    int nThreads = BATCH * HN * HW;                  // 401408
    softmax_agg_kernel<<<nThreads / 256, 256, 0, stream>>>(lgw, x3w, out);
  }
}